// RelGraphConvHomo_52690658787906
// MI455X (gfx1250) — compile-verified
//
#include <hip/hip_runtime.h>
#include <hip/hip_bf16.h>

typedef __attribute__((ext_vector_type(2))) float v2f;
typedef __attribute__((ext_vector_type(8))) float v8f;

#define IN_FEAT  64
#define OUT_FEAT 64
#define N_RELS   16

__device__ __forceinline__ void atomAddF(float* p, float v) {
    __hip_atomic_fetch_add(p, v, __ATOMIC_RELAXED, __HIP_MEMORY_SCOPE_AGENT);
}

// ---------------------------------------------------------------------------
// Zero the output accumulator.
// ---------------------------------------------------------------------------
__global__ void rgcn_zero(float* __restrict__ p, long n) {
    long i = (long)blockIdx.x * blockDim.x + threadIdx.x;
    if (i < n) p[i] = 0.0f;
}

// ---------------------------------------------------------------------------
// Phase 1: hx[rc][n][o] = sum_i feat[n][i] * W[r0+rc][o][i]
// One wave computes a 16-row x 64-col strip via 4 accumulators of
// V_WMMA_F32_16X16X4_F32 (exact fp32 on the matrix pipe), K-loop of 16 steps.
//
// f32 16x16x4 layouts (ISA 7.12.2):
//   A 16x4 : lane L holds A[M=L%16][K = 2*(L/16) + p] in VGPR p (p=0,1)
//   B 4x16 : lane L holds B[K = p + 2*(L/16)][N=L%16]  in VGPR p
//   C 16x16: lane L VGPR j holds C[M = j + 8*(L/16)][N = L%16]
// ---------------------------------------------------------------------------
__global__ void rgcn_gemm(const float* __restrict__ feat,
                          const float* __restrict__ W,
                          float* __restrict__ hx,
                          int r0, int N, int mtiles) {
    const int lane  = threadIdx.x & 31;
    const int wid   = threadIdx.x >> 5;
    const int mtile = blockIdx.x * (blockDim.x >> 5) + wid;
    if (mtile >= mtiles) return;

    const int rc = blockIdx.y;
    const int r  = r0 + rc;

    const int m16   = lane & 15;
    const int khalf = (lane >> 4) << 1;          // 0 or 2
    int mrow = mtile * 16 + m16;
    if (mrow >= N) mrow = N - 1;                 // clamp (stores are guarded)

    const float* featRow = feat + (size_t)mrow * IN_FEAT;
    const float* Wr      = W + (size_t)r * (OUT_FEAT * IN_FEAT);

    v8f acc0 = {}, acc1 = {}, acc2 = {}, acc3 = {};

    for (int k = 0; k < IN_FEAT / 4; ++k) {
        const int kk = k * 4 + khalf;
        v2f a  = *(const v2f*)(featRow + kk);
        // B[k'][o] = W[r][o][k'] ; per-lane o = nt*16 + m16, elems kk, kk+1
        v2f b0 = *(const v2f*)(Wr + (size_t)(m16 +  0) * IN_FEAT + kk);
        v2f b1 = *(const v2f*)(Wr + (size_t)(m16 + 16) * IN_FEAT + kk);
        v2f b2 = *(const v2f*)(Wr + (size_t)(m16 + 32) * IN_FEAT + kk);
        v2f b3 = *(const v2f*)(Wr + (size_t)(m16 + 48) * IN_FEAT + kk);
        acc0 = __builtin_amdgcn_wmma_f32_16x16x4_f32(false, a, false, b0, (short)0, acc0, false, false);
        acc1 = __builtin_amdgcn_wmma_f32_16x16x4_f32(false, a, false, b1, (short)0, acc1, false, false);
        acc2 = __builtin_amdgcn_wmma_f32_16x16x4_f32(false, a, false, b2, (short)0, acc2, false, false);
        acc3 = __builtin_amdgcn_wmma_f32_16x16x4_f32(false, a, false, b3, (short)0, acc3, false, false);
    }

    const int col   = lane & 15;
    const int rbase = mtile * 16 + ((lane >> 4) << 3);
    float* base = hx + (size_t)rc * N * OUT_FEAT;
    for (int j = 0; j < 8; ++j) {
        const int row = rbase + j;
        if (row < N) {
            float* o = base + (size_t)row * OUT_FEAT + col;
            o[0]  = acc0[j];
            o[16] = acc1[j];
            o[32] = acc2[j];
            o[48] = acc3[j];
        }
    }
}

// ---------------------------------------------------------------------------
// Phase 2: edge sweep. One wave scans a contiguous window of edges (dst is
// sorted), register-accumulating 64 outputs (2/lane) while dst is unchanged;
// flushes one fp32-atomic burst per run boundary. Edges whose etype is
// outside [r0,r1) are skipped but still advance the run tracking.
// hx rows are L2-resident (<=192MB chunk), so the random gather hits L2.
// ---------------------------------------------------------------------------
__global__ void rgcn_scatter(const float* __restrict__ hx,
                             const int* __restrict__ src,
                             const int* __restrict__ dst,
                             const int* __restrict__ et,
                             float* __restrict__ Y,
                             int r0, int r1, int nEdges, int edgesPerWave,
                             int N) {
    const int wave = (blockIdx.x * blockDim.x + threadIdx.x) >> 5;
    const int lane = threadIdx.x & 31;
    long e0 = (long)wave * edgesPerWave;
    if (e0 >= nEdges) return;
    long e1 = e0 + edgesPerWave;
    if (e1 > nEdges) e1 = nEdges;

    float a0 = 0.0f, a1 = 0.0f;
    int cur = dst[e0];

    for (long e = e0; e < e1; ++e) {
        const int d = dst[e];
        if (d != cur) {
            if (a0 != 0.0f) atomAddF(&Y[(size_t)cur * OUT_FEAT + lane], a0);
            if (a1 != 0.0f) atomAddF(&Y[(size_t)cur * OUT_FEAT + lane + 32], a1);
            a0 = 0.0f; a1 = 0.0f;
            cur = d;
        }
        const int r = et[e];
        if (r >= r0 && r < r1) {
            const float* row = hx +
                ((size_t)(r - r0) * N + (size_t)src[e]) * OUT_FEAT;
            a0 += row[lane];
            a1 += row[lane + 32];
        }
    }
    if (a0 != 0.0f) atomAddF(&Y[(size_t)cur * OUT_FEAT + lane], a0);
    if (a1 != 0.0f) atomAddF(&Y[(size_t)cur * OUT_FEAT + lane + 32], a1);
}

// ---------------------------------------------------------------------------
// Fallback if ws can't hold even one relation slice: direct per-edge matvec.
// ---------------------------------------------------------------------------
__global__ void rgcn_direct(const float* __restrict__ feat,
                            const float* __restrict__ W,
                            const int* __restrict__ src,
                            const int* __restrict__ dst,
                            const int* __restrict__ et,
                            float* __restrict__ Y, int nEdges) {
    const int e    = (blockIdx.x * blockDim.x + threadIdx.x) >> 5;
    const int lane = threadIdx.x & 31;
    if (e >= nEdges) return;
    const int s = src[e], d = dst[e], r = et[e];
    const float* x  = feat + (size_t)s * IN_FEAT;
    const float* w0 = W + (size_t)r * (OUT_FEAT * IN_FEAT) + (size_t)lane * IN_FEAT;
    const float* w1 = w0 + 32 * IN_FEAT;
    float a0 = 0.0f, a1 = 0.0f;
    for (int i = 0; i < IN_FEAT; ++i) {
        const float xv = x[i];
        a0 += w0[i] * xv;
        a1 += w1[i] * xv;
    }
    atomAddF(&Y[(size_t)d * OUT_FEAT + lane], a0);
    atomAddF(&Y[(size_t)d * OUT_FEAT + lane + 32], a1);
}

extern "C" void kernel_launch(void* const* d_in, const int* in_sizes, int n_in,
                              void* d_out, int out_size, void* d_ws, size_t ws_size,
                              hipStream_t stream) {
    const float* feat = (const float*)d_in[0];
    const float* W    = (const float*)d_in[1];
    const int*   src  = (const int*)d_in[2];
    const int*   dst  = (const int*)d_in[3];
    const int*   et   = (const int*)d_in[4];
    float* Y = (float*)d_out;

    const int N = in_sizes[0] / IN_FEAT;        // 50000
    const int E = in_sizes[2];                  // 1,600,000
    const int R = in_sizes[1] / (OUT_FEAT * IN_FEAT); // 16

    // Zero the accumulator output.
    {
        long n = out_size;
        int blocks = (int)((n + 255) / 256);
        rgcn_zero<<<blocks, 256, 0, stream>>>(Y, n);
    }

    const size_t perRel = (size_t)N * OUT_FEAT * sizeof(float); // 12.8 MB
    int RC = (ws_size >= perRel) ? (int)(ws_size / perRel) : 0;
    if (RC > R) RC = R;

    if (RC < 1) {
        // No usable workspace: direct per-edge matvec fallback.
        const int waves  = E;
        const int blocks = (waves + 7) / 8;     // 8 waves / 256-thread block
        rgcn_direct<<<blocks, 256, 0, stream>>>(feat, W, src, dst, et, Y, E);
        return;
    }

    float* hx = (float*)d_ws;
    const int mtiles     = (N + 15) / 16;
    const int gemmBlocks = (mtiles + 7) / 8;

    const int EPW         = 64;                 // edges per wave in the sweep
    const int sweepWaves  = (E + EPW - 1) / EPW;
    const int sweepBlocks = (sweepWaves + 7) / 8;

    for (int r0 = 0; r0 < R; r0 += RC) {
        int rc = R - r0;
        if (rc > RC) rc = RC;
        rgcn_gemm<<<dim3(gemmBlocks, rc), 256, 0, stream>>>(feat, W, hx, r0, N, mtiles);
        rgcn_scatter<<<sweepBlocks, 256, 0, stream>>>(hx, src, dst, et, Y,
                                                      r0, r0 + rc, E, EPW, N);
    }
}